// DiffGCN_63041529970994
// MI455X (gfx1250) — compile-verified
//
#include <hip/hip_runtime.h>
#include <hip/hip_bf16.h>
#include <math.h>

typedef __attribute__((ext_vector_type(2))) float v2f;
typedef __attribute__((ext_vector_type(8))) float v8f;

#define D_EMB 64
#define K_NBR 32
#define T_STEPS 3
#define EPS_NOISE 0.01f

// ---------------------------------------------------------------------------
// Phase 1: S[n][t] = dot(v[n], W[t*64 : (t+1)*64])  via V_WMMA_F32_16X16X4_F32
// One wave computes a 16-row tile. A = 16x4 f32 slab of v (ISA layout: lanes
// 0-15 hold {K0,K1}, lanes 16-31 hold {K2,K3} for row = base + (lane&15) --
// exactly a contiguous float2 load). B = 4x16 with N-columns 0..2 = W chunks.
//
// Branchless trick: WMMA columns are independent (D[:,n] = A @ B[:,n]), so
// unused B columns 3..15 may hold garbage -- we clamp the column index and
// load duplicate (valid) W data unconditionally instead of masking to zero.
// Only D columns 0..2 are stored. Inner loop = 2 loads + 1 wmma, no branches.
// ---------------------------------------------------------------------------
__global__ __launch_bounds__(256)
void diffgcn_scores_wmma(const float* __restrict__ v,
                         const float* __restrict__ W,
                         float* __restrict__ S,
                         int N) {
    const int lane = threadIdx.x & 31;
    const int wave = (blockIdx.x * blockDim.x + threadIdx.x) >> 5;
    const int base = wave * 16;                 // 16 rows per wave
    if (base >= N) return;                      // N % 16 == 0 for N=200000

    const int col   = lane & 15;                // N-column this lane owns
    const int colc  = (col < T_STEPS) ? col : (T_STEPS - 1);  // clamp: dup col2
    const int khalf = (lane >> 4) * 2;          // 0 for lanes 0-15, 2 for 16-31

    const float* __restrict__ vp = v + (size_t)(base + col) * D_EMB + khalf;
    const float* __restrict__ wp = W + colc * D_EMB + khalf;

    v8f acc = {};
    #pragma unroll
    for (int k0 = 0; k0 < D_EMB; k0 += 4) {
        v2f a = *(const v2f*)(vp + k0);         // A frag: {K0,K1}/{K2,K3}
        v2f b = *(const v2f*)(wp + k0);         // B frag (cols>=3: dup of col2)
        acc = __builtin_amdgcn_wmma_f32_16x16x4_f32(
            /*neg_a=*/false, a, /*neg_b=*/false, b,
            /*c_mod=*/(short)0, acc, /*reuse_a=*/false, /*reuse_b=*/false);
    }

    // D layout: VGPR r, lanes 0-15 -> (M=r, N=lane); lanes 16-31 -> (M=8+r).
    if (col < T_STEPS) {
        const int mbase = base + ((lane >> 4) ? 8 : 0);
        #pragma unroll
        for (int r = 0; r < 8; ++r) {
            S[(size_t)(mbase + r) * T_STEPS + col] = acc[r];
        }
    }
}

// ---------------------------------------------------------------------------
// Deterministic stand-in for jax.random.normal (threefry is not reproducible
// here): PCG hash -> Box-Muller.
// ---------------------------------------------------------------------------
__device__ __forceinline__ unsigned pcg_hash(unsigned x) {
    x = x * 747796405u + 2891336453u;
    unsigned w = ((x >> ((x >> 28u) + 4u)) ^ x) * 277803737u;
    return (w >> 22u) ^ w;
}

__device__ __forceinline__ float gauss_noise(unsigned seed) {
    unsigned a = pcg_hash(seed);
    unsigned b = pcg_hash(seed ^ 0x9E3779B9u);
    float u1 = (float)(a >> 8) * (1.0f / 16777216.0f) + 1.0e-7f;
    float u2 = (float)(b >> 8) * (1.0f / 16777216.0f);
    return __builtin_sqrtf(-2.0f * __logf(u1)) * __cosf(6.2831853071f * u2);
}

__device__ __forceinline__ float wave_max32(float x) {
    #pragma unroll
    for (int o = 16; o > 0; o >>= 1) x = fmaxf(x, __shfl_xor(x, o, 32));
    return x;
}
__device__ __forceinline__ float wave_sum32(float x) {
    #pragma unroll
    for (int o = 16; o > 0; o >>= 1) x += __shfl_xor(x, o, 32);
    return x;
}

// ---------------------------------------------------------------------------
// Phase 2: one wave32 per walk, one lane per candidate. All T=3 steps in one
// kernel (the recurrence is per-walk independent).  base-term dropped: softmax
// is shift-invariant per row, so p = softmax(s[cand]).
// ---------------------------------------------------------------------------
__global__ __launch_bounds__(256)
void diffgcn_walk(const float* __restrict__ v,
                  const int*   __restrict__ nbr,
                  const float* __restrict__ S,
                  int*         __restrict__ walks_out,   // (N, T+1) int32
                  float*       __restrict__ embeds_out,  // (N, T*64) f32
                  int N) {
    const int lane = threadIdx.x & 31;
    const int n    = (blockIdx.x * blockDim.x + threadIdx.x) >> 5;  // walk id
    if (n >= N) return;

    int cur = n;                                  // walks[:,0] = arange(N)
    if (lane == 0) walks_out[(size_t)n * (T_STEPS + 1)] = n;

    #pragma unroll
    for (int t = 0; t < T_STEPS; ++t) {
        // candidate for this lane (coalesced 128B row of nbr)
        int c = nbr[(size_t)cur * K_NBR + lane];
        // gathered score (S table is 2.4MB -> L2 resident)
        float s = S[(size_t)c * T_STEPS + t];

        // softmax over the 32 candidates (wave32 butterflies)
        float m = wave_max32(s);
        float e = __expf(s - m);
        float sum = wave_sum32(e);
        float p = e / sum;

        // noisy argmax; tie-break = lowest index (matches jnp.argmax)
        unsigned seed = ((unsigned)t * 0x01000193u + (unsigned)n) * 32u
                        + (unsigned)lane;
        float noisy = p + EPS_NOISE * gauss_noise(seed);

        float best = noisy;
        int   bidx = lane;
        int   bc   = c;
        #pragma unroll
        for (int o = 16; o > 0; o >>= 1) {
            float ov = __shfl_xor(best, o, 32);
            int   oi = __shfl_xor(bidx, o, 32);
            int   oc = __shfl_xor(bc,   o, 32);
            if (ov > best || (ov == best && oi < bidx)) {
                best = ov; bidx = oi; bc = oc;
            }
        }
        cur = bc;                                 // all lanes agree

        if (lane == 0) walks_out[(size_t)n * (T_STEPS + 1) + t + 1] = cur;

        // walk_embeds[:, t*64:(t+1)*64] = v[cur]  (slice written exactly once)
        const v2f* src = (const v2f*)(v + (size_t)cur * D_EMB) + lane;
        v2f* dst = (v2f*)(embeds_out + (size_t)n * (T_STEPS * D_EMB)
                          + t * D_EMB) + lane;
        *dst = *src;
    }
}

extern "C" void kernel_launch(void* const* d_in, const int* in_sizes, int n_in,
                              void* d_out, int out_size, void* d_ws, size_t ws_size,
                              hipStream_t stream) {
    const float* v   = (const float*)d_in[0];   // (N, 64) f32
    const int*   nbr = (const int*)  d_in[1];   // (N, 32) i32
    const float* W   = (const float*)d_in[2];   // (192, 1) f32
    // d_in[3] = b (dead: softmax shift-invariance)

    const int N = in_sizes[0] / D_EMB;

    float* S = (float*)d_ws;                    // (N, 3) scores scratch

    // d_out = [walks (N x 4) int32 bits | walk_embeds (N x 192) f32]
    int*   walks_out  = (int*)d_out;
    float* embeds_out = (float*)d_out + (size_t)N * (T_STEPS + 1);

    // Phase 1: 1 wave per 16 rows, 8 waves (256 thr) per block.
    {
        int waves  = (N + 15) / 16;
        int blocks = (waves + 7) / 8;
        diffgcn_scores_wmma<<<blocks, 256, 0, stream>>>(v, W, S, N);
    }
    // Phase 2: 1 wave per walk, 8 walks per 256-thread block.
    {
        int blocks = (N + 7) / 8;
        diffgcn_walk<<<blocks, 256, 0, stream>>>(v, nbr, S, walks_out,
                                                 embeds_out, N);
    }
}